// ClimbGNN_38517266711198
// MI455X (gfx1250) — compile-verified
//
#include <hip/hip_runtime.h>

// ---------------------------------------------------------------------------
// 2-layer GCN for MI455X (gfx1250, wave32).
//   deg/dinv  -> fp32 WMMA dense transform (V_WMMA_F32_16X16X4_F32)
//   -> per-edge scatter-add via global_atomic_add_f32 (L2-resident, 192MB L2)
//   -> finalize (self-loop folded in, bias, relu)
// ---------------------------------------------------------------------------

typedef __attribute__((ext_vector_type(2))) float v2f;
typedef __attribute__((ext_vector_type(8))) float v8f;

__device__ __forceinline__ v8f wmma_f32_16x16x4(v2f a, v2f b, v8f c) {
  // 8 args: (neg_a, A, neg_b, B, c_mod, C, reuse_a, reuse_b)
  return __builtin_amdgcn_wmma_f32_16x16x4_f32(false, a, false, b, (short)0, c,
                                               false, false);
}

// ---- degree / normalization ------------------------------------------------

__global__ void k_deg_init(float* __restrict__ deg, int n) {
  int i = blockIdx.x * blockDim.x + threadIdx.x;
  if (i < n) deg[i] = 1.0f;  // self-loop contributes 1 to every node's degree
}

__global__ void k_deg_count(const long long* __restrict__ ei,
                            float* __restrict__ deg, long long E) {
  long long e = (long long)blockIdx.x * blockDim.x + threadIdx.x;
  if (e < E) {
    int d = (int)ei[E + e];  // dst row of edge_index
    __hip_atomic_fetch_add(&deg[d], 1.0f, __ATOMIC_RELAXED,
                           __HIP_MEMORY_SCOPE_AGENT);
  }
}

__global__ void k_rsqrt(float* __restrict__ deg, int n) {
  int i = blockIdx.x * blockDim.x + threadIdx.x;
  if (i < n) deg[i] = rsqrtf(deg[i]);  // deg >= 1, in-place -> dinv
}

// ---- dense transform: M[i,:] = dinv[i] * (X[i,:] @ W), agg[i,:] = 0 --------
// One wave computes a 16-row x 32-col output tile using fp32 WMMA.
// K in {16, 32}. W staged in LDS. EXEC all-ones at every WMMA (wave-uniform
// tile guard).

template <int K>
__global__ void __launch_bounds__(256) k_transform(
    const float* __restrict__ X, const float* __restrict__ W,
    const float* __restrict__ dinv, float* __restrict__ M,
    float* __restrict__ agg, int nTiles) {
  __shared__ float sW[K * 32];
  for (int i = threadIdx.x; i < K * 32; i += blockDim.x) sW[i] = W[i];
  __syncthreads();

  const int wave = threadIdx.x >> 5;
  const int lane = threadIdx.x & 31;
  const int tile = blockIdx.x * 8 + wave;
  if (tile >= nTiles) return;  // wave-uniform branch

  const int rowBase = tile * 16;
  const int mrow = rowBase + (lane & 15);   // A: M index
  const int khalf = (lane >> 4) * 2;        // A/B: lanes 16-31 hold K+2,K+3
  const int col = lane & 15;                // B/C/D: N index

  v8f c0 = {};  // output cols [0,16)
  v8f c1 = {};  // output cols [16,32)
#pragma unroll
  for (int k0 = 0; k0 < K; k0 += 4) {
    // A fragment: {X[mrow][k0+khalf], X[mrow][k0+khalf+1]} (8B aligned)
    v2f a = *(const v2f*)(X + (size_t)mrow * K + k0 + khalf);
    v2f b0, b1;
    b0.x = sW[(k0 + khalf) * 32 + col];
    b0.y = sW[(k0 + khalf + 1) * 32 + col];
    b1.x = sW[(k0 + khalf) * 32 + 16 + col];
    b1.y = sW[(k0 + khalf + 1) * 32 + 16 + col];
    c0 = wmma_f32_16x16x4(a, b0, c0);
    c1 = wmma_f32_16x16x4(a, b1, c1);
  }

  // C/D layout: lane L, VGPR r -> row = rowBase + (L>>4)*8 + r, col = n0+(L&15)
  const int rhalf = (lane >> 4) * 8;
#pragma unroll
  for (int r = 0; r < 8; ++r) {
    int row = rowBase + rhalf + r;
    float s = dinv[row];  // pre-scale message by dinv[src]
    size_t o = (size_t)row * 32 + col;
    M[o] = s * c0[r];
    M[o + 16] = s * c1[r];
    agg[o] = 0.0f;
    agg[o + 16] = 0.0f;
  }
}

// ---- edge scatter: agg[dst,:] += M[src,:] ----------------------------------
// One wave per edge: indices are wave-uniform -> readfirstlane to SGPRs,
// each lane handles one of the 32 features (coalesced 128B gather + atomics).

__global__ void __launch_bounds__(256) k_scatter(
    const long long* __restrict__ ei, const float* __restrict__ M,
    float* __restrict__ agg, long long E) {
  long long e = (long long)blockIdx.x * 8 + (threadIdx.x >> 5);
  int f = threadIdx.x & 31;
  if (e < E) {  // wave-uniform
    int s = __builtin_amdgcn_readfirstlane((int)ei[e]);
    int d = __builtin_amdgcn_readfirstlane((int)ei[E + e]);
    float v = M[(size_t)s * 32 + f];
    __hip_atomic_fetch_add(&agg[(size_t)d * 32 + f], v, __ATOMIC_RELAXED,
                           __HIP_MEMORY_SCOPE_AGENT);
  }
}

// ---- finalize: out = dinv[i]*(agg[i,:] + M[i,:]) + b  (+relu) --------------
// The + M[i,:] term is the self-loop (already scaled by dinv[i] once).

template <bool RELU>
__global__ void k_finalize(const float* __restrict__ agg,
                           const float* __restrict__ M,
                           const float* __restrict__ dinv,
                           const float* __restrict__ bias,
                           float* __restrict__ out, long long total) {
  long long t = (long long)blockIdx.x * blockDim.x + threadIdx.x;
  if (t < total) {
    int row = (int)(t >> 5);
    int col = (int)(t & 31);
    float v = dinv[row] * (agg[t] + M[t]) + bias[col];
    if (RELU) v = fmaxf(v, 0.0f);
    out[t] = v;
  }
}

// ---------------------------------------------------------------------------

extern "C" void kernel_launch(void* const* d_in, const int* in_sizes, int n_in,
                              void* d_out, int out_size, void* d_ws,
                              size_t ws_size, hipStream_t stream) {
  const float* x = (const float*)d_in[0];
  const long long* ei = (const long long*)d_in[1];  // int64 [2, E]
  const float* W1 = (const float*)d_in[2];
  const float* b1 = (const float*)d_in[3];
  const float* W2 = (const float*)d_in[4];
  const float* b2 = (const float*)d_in[5];
  float* out = (float*)d_out;

  const int N = in_sizes[0] / 16;            // 200000
  const long long E = in_sizes[1] / 2;       // 2400000
  const int nTiles = (N + 15) / 16;          // 12500

  // Workspace carve-out (all 256B aligned): dinv | M | agg | h  (~78 MB)
  char* ws = (char*)d_ws;
  size_t offM = ((size_t)N * 4 + 255) & ~(size_t)255;
  size_t feat = (size_t)N * 32 * 4;
  float* dinv = (float*)ws;
  float* M = (float*)(ws + offM);
  float* agg = (float*)(ws + offM + feat);
  float* h = (float*)(ws + offM + 2 * feat);

  const int B = 256;
  const int gN = (N + B - 1) / B;
  const int gE = (int)((E + B - 1) / B);
  const int gT = (nTiles + 7) / 8;               // 8 waves (tiles) per block
  const int gS = (int)((E + 7) / 8);             // 1 wave per edge
  const long long total = (long long)N * 32;
  const int gF = (int)((total + B - 1) / B);

  // normalization
  k_deg_init<<<gN, B, 0, stream>>>(dinv, N);
  k_deg_count<<<gE, B, 0, stream>>>(ei, dinv, E);
  k_rsqrt<<<gN, B, 0, stream>>>(dinv, N);

  // layer 1: m' = dinv*(x@W1); agg += over edges; h = relu(dinv*(agg+m')+b1)
  k_transform<16><<<gT, B, 0, stream>>>(x, W1, dinv, M, agg, nTiles);
  k_scatter<<<gS, B, 0, stream>>>(ei, M, agg, E);
  k_finalize<true><<<gF, B, 0, stream>>>(agg, M, dinv, b1, h, total);

  // layer 2: same with W2/b2, no relu, straight to d_out
  k_transform<32><<<gT, B, 0, stream>>>(h, W2, dinv, M, agg, nTiles);
  k_scatter<<<gS, B, 0, stream>>>(ei, M, agg, E);
  k_finalize<false><<<gF, B, 0, stream>>>(agg, M, dinv, b2, out, total);
}